// SPAIRGridFeatureNetwork_15470472200207
// MI455X (gfx1250) — compile-verified
//
#include <hip/hip_runtime.h>
#include <math.h>

typedef float v2f __attribute__((ext_vector_type(2)));
typedef float v8f __attribute__((ext_vector_type(8)));

#define ACT_NONE 0
#define ACT_RELU 1
#define ACT_CELU 2

__device__ __forceinline__ v8f wmma4(v2f a, v2f b, v8f c) {
  // D = A(16x4,f32) * B(4x16,f32) + C(16x16,f32)
  return __builtin_amdgcn_wmma_f32_16x16x4_f32(false, a, false, b, (short)0, c,
                                               false, false);
}

__device__ __forceinline__ float apply_act(float v, int act) {
  if (act == ACT_RELU) return fmaxf(v, 0.f);
  if (act == ACT_CELU) return v > 0.f ? v : expm1f(v);
  return v;
}

__device__ __forceinline__ void atomicMaxF(float* p, float v) {
  __hip_atomic_fetch_max(p, v, __ATOMIC_RELAXED, __HIP_MEMORY_SCOPE_AGENT);
}

// ---------------------------------------------------------------------------
// Generic WMMA GEMM: Y(N,M) = act( X(N,K) @ W(K,M) + bias + resid )
// One wave computes a 16x16 tile. Branch-free inner loop:
//  - out-of-range rows/cols are address-CLAMPED (loads always legal),
//    and masked only at the store.
//  - K tail (K%4 != 0) handled by one extra step with A value-masked to 0
//    and B address-clamped (zero A annihilates the product).
// ---------------------------------------------------------------------------
__global__ __launch_bounds__(32)
void gemm16(const float* __restrict__ X, const float* __restrict__ W,
            const float* __restrict__ bias, const float* __restrict__ resid,
            float* __restrict__ Y, int N, int K, int M, int act)
{
  const int lane = threadIdx.x & 31;
  const int m = lane & 15;
  const int hi = lane >> 4;
  const int col = blockIdx.x * 16 + m;
  const int rowBase = blockIdx.y * 16;
  int rowA = rowBase + m;
  if (rowA > N - 1) rowA = N - 1;                 // clamp; masked at store
  const bool colOK = (col < M);
  const int colC = colOK ? col : (M - 1);          // clamp; masked at store
  const float* __restrict__ Xr = X + (size_t)rowA * K;
  const float* __restrict__ Wc = W + colC;

  v8f acc = {0.f, 0.f, 0.f, 0.f, 0.f, 0.f, 0.f, 0.f};
  const int Kmain = K & ~3;
  int k0 = 0;
#pragma unroll 2
  for (; k0 < Kmain; k0 += 4) {
    const int ka = k0 + 2 * hi;
    v2f a, b;
    a.x = Xr[ka];
    a.y = Xr[ka + 1];
    b.x = Wc[(size_t)ka * M];
    b.y = Wc[(size_t)(ka + 1) * M];
    acc = wmma4(a, b, acc);
  }
  if (k0 < K) {                                    // tail: K%4 in {1,2,3}
    const int ka = k0 + 2 * hi;
    const int kb0 = (ka     < K) ? ka     : (K - 1);
    const int kb1 = (ka + 1 < K) ? ka + 1 : (K - 1);
    v2f a, b;
    a.x = (ka     < K) ? Xr[ka]     : 0.f;
    a.y = (ka + 1 < K) ? Xr[ka + 1] : 0.f;
    b.x = Wc[(size_t)kb0 * M];
    b.y = Wc[(size_t)kb1 * M];
    acc = wmma4(a, b, acc);
  }

  const float bv = (bias != nullptr && colOK) ? bias[col] : 0.f;
#pragma unroll
  for (int r = 0; r < 8; ++r) {
    const int row = rowBase + r + 8 * hi;
    if (row < N && colOK) {
      float v = acc[r] + bv;
      if (resid) v += resid[(size_t)row * M + col];
      Y[(size_t)row * M + col] = apply_act(v, act);
    }
  }
}

// ---------------------------------------------------------------------------
// Fused per-edge MLP for the auto-registration layer:
//   e   = [pos[src]-pos[tgt]-dp[tgt], x[src]]     (16 edges staged in LDS,
//                                                  zero-padded to stride d+4)
//   hid = relu(e @ wf1 + bf1)                     (LDS, padded stride)
//   out = hid @ wf2 + bf2
//   agg[tgt] = max(agg[tgt], out)                 (native float atomic max)
// blockDim.x = 2*d  (d/16 waves, one 16-col tile each). d % 4 == 0.
// ---------------------------------------------------------------------------
__global__ __launch_bounds__(512)
void ar_edge(const float* __restrict__ x, const float* __restrict__ pos,
             const float* __restrict__ dp,
             const int* __restrict__ esrc, const int* __restrict__ etgt, int E,
             const float* __restrict__ wf1, const float* __restrict__ bf1,
             const float* __restrict__ wf2, const float* __restrict__ bf2,
             float* __restrict__ agg, int d)
{
  extern __shared__ float smem[];
  const int K1 = d + 3;
  const int K1p = d + 4;          // padded LDS stride (col K1 zero-padded)
  const int Hp = d + 4;
  float* As  = smem;              // 16 * K1p
  float* Hid = smem + 16 * K1p;   // 16 * Hp
  __shared__ int s_src[16];
  __shared__ int s_tgt[16];

  const int tid = threadIdx.x;
  const int lane = tid & 31;
  const int wave = tid >> 5;
  const int ebase = blockIdx.x * 16;

  if (tid < 16) {
    int e = ebase + tid;
    if (e > E - 1) e = E - 1;
    s_src[tid] = esrc[e];
    s_tgt[tid] = etgt[e];
  }
  __syncthreads();

  // Stage gathered A tile (16 x K1p, zero-padded) into LDS.
  for (int i = tid; i < 16 * K1p; i += blockDim.x) {
    const int mr = i / K1p;
    const int k = i - mr * K1p;
    const int s = s_src[mr];
    const int t = s_tgt[mr];
    float v = 0.f;
    if (k < 3)        v = pos[s * 3 + k] - pos[t * 3 + k] - dp[t * 3 + k];
    else if (k < K1)  v = x[(size_t)s * d + (k - 3)];
    As[i] = v;
  }
  __syncthreads();

  const int m = lane & 15;
  const int hi = lane >> 4;
  const int col = wave * 16 + m;   // always < d

  // Layer 1: hid = relu(A @ wf1 + bf1). Main loop k0 < d is branch-free;
  // one tail step covers k = d..d+2 (A padded zero at d+3, wf1 row clamped).
  v8f acc = {0.f, 0.f, 0.f, 0.f, 0.f, 0.f, 0.f, 0.f};
  const float* __restrict__ AsR = As + m * K1p;
  const float* __restrict__ W1c = wf1 + col;
#pragma unroll 2
  for (int k0 = 0; k0 < d; k0 += 4) {
    const int ka = k0 + 2 * hi;
    v2f a, b;
    a.x = AsR[ka];
    a.y = AsR[ka + 1];
    b.x = W1c[(size_t)ka * d];
    b.y = W1c[(size_t)(ka + 1) * d];
    acc = wmma4(a, b, acc);
  }
  {
    const int ka = d + 2 * hi;                 // ka <= d+2 == K1-1, valid
    const int kb1 = (ka + 1 < K1) ? ka + 1 : (K1 - 1);
    v2f a, b;
    a.x = AsR[ka];
    a.y = AsR[ka + 1];                         // zero-padded at k == d+3
    b.x = W1c[(size_t)ka * d];
    b.y = W1c[(size_t)kb1 * d];
    acc = wmma4(a, b, acc);
  }
  {
    const float bv = bf1[col];
#pragma unroll
    for (int r = 0; r < 8; ++r) {
      const int rr = r + 8 * hi;
      Hid[rr * Hp + col] = fmaxf(acc[r] + bv, 0.f);
    }
  }
  __syncthreads();

  // Layer 2: out = hid @ wf2 + bf2 ; scatter-max into agg[tgt]
  v8f acc2 = {0.f, 0.f, 0.f, 0.f, 0.f, 0.f, 0.f, 0.f};
  const float* __restrict__ HidR = Hid + m * Hp;
  const float* __restrict__ W2c = wf2 + col;
#pragma unroll 2
  for (int k0 = 0; k0 < d; k0 += 4) {
    const int ka = k0 + 2 * hi;
    v2f a, b;
    a.x = HidR[ka];
    a.y = HidR[ka + 1];
    b.x = W2c[(size_t)ka * d];
    b.y = W2c[(size_t)(ka + 1) * d];
    acc2 = wmma4(a, b, acc2);
  }
  const float bv2 = bf2[col];
#pragma unroll
  for (int r = 0; r < 8; ++r) {
    const int rr = r + 8 * hi;
    const int e = ebase + rr;
    if (e < E) {
      atomicMaxF(&agg[(size_t)s_tgt[rr] * d + col], acc2[r] + bv2);
    }
  }
}

// ---------------------------------------------------------------------------
// Small helper kernels
// ---------------------------------------------------------------------------
__global__ void fill_f32(float* __restrict__ p, float v, long n) {
  long i = (long)blockIdx.x * blockDim.x + threadIdx.x;
  if (i < n) p[i] = v;
}

__global__ void fill_i32(int* __restrict__ p, int v, long n) {
  long i = (long)blockIdx.x * blockDim.x + threadIdx.x;
  if (i < n) p[i] = v;
}

__global__ void build_rel(const float* __restrict__ pin,
                          const float* __restrict__ pout,
                          const int* __restrict__ idx,
                          float* __restrict__ rel, int N)
{
  int i = blockIdx.x * blockDim.x + threadIdx.x;
  if (i >= N * 3) return;
  int row = i / 3, k = i - row * 3;
  rel[i] = pin[i] - pout[idx[row] * 3 + k];
}

__global__ void scatter_hw(const float* __restrict__ h, const float* __restrict__ w,
                           const int* __restrict__ idx,
                           float* __restrict__ sum, int* __restrict__ cnt,
                           long NC, int C)
{
  long i = (long)blockIdx.x * blockDim.x + threadIdx.x;
  if (i >= NC) return;
  int row = (int)(i / C);
  int c = (int)(i - (long)row * C);
  int o = idx[row];
  atomicAdd(&sum[(size_t)o * C + c], h[i] * w[i]);
  if (c == 0) atomicAdd(&cnt[o], 1);
}

__global__ void div_cnt(float* __restrict__ sum, const int* __restrict__ cnt,
                        long MC, int C)
{
  long i = (long)blockIdx.x * blockDim.x + threadIdx.x;
  if (i >= MC) return;
  int row = (int)(i / C);
  int c = cnt[row];
  sum[i] /= (float)(c < 1 ? 1 : c);
}

// Graph-norm hierarchical reduction (B == 2).
// mode 0: gsum[b,c] += x ; gcnt[b] += 1 per row
// mode 1: gsum[b,c] += (x - alpha[c]*mean[b,c])^2
__global__ __launch_bounds__(256)
void gnorm_accum(const float* __restrict__ x, const int* __restrict__ bidx,
                 const float* __restrict__ alpha, const float* __restrict__ mean,
                 float* __restrict__ gsum, int* __restrict__ gcnt,
                 int N, int C, int mode)
{
  __shared__ float ls[512];
  __shared__ int lc[2];
  for (int i = threadIdx.x; i < 2 * C; i += blockDim.x) ls[i] = 0.f;
  if (threadIdx.x < 2) lc[threadIdx.x] = 0;
  __syncthreads();
  const long total = (long)N * C;
  for (long i = (long)blockIdx.x * blockDim.x + threadIdx.x; i < total;
       i += (long)gridDim.x * blockDim.x) {
    int row = (int)(i / C);
    int c = (int)(i - (long)row * C);
    int b = bidx[row];
    float v = x[i];
    if (mode == 1) {
      float s = v - alpha[c] * mean[b * C + c];
      v = s * s;
    }
    atomicAdd(&ls[b * C + c], v);
    if (mode == 0 && c == 0) atomicAdd(&lc[b], 1);
  }
  __syncthreads();
  for (int i = threadIdx.x; i < 2 * C; i += blockDim.x) atomicAdd(&gsum[i], ls[i]);
  if (mode == 0 && threadIdx.x < 2) atomicAdd(&gcnt[threadIdx.x], lc[threadIdx.x]);
}

__global__ void gnorm_finalize(float* __restrict__ v, const int* __restrict__ cnt, int C)
{
  int i = blockIdx.x * blockDim.x + threadIdx.x;
  if (i >= 2 * C) return;
  int b = i / C;
  int c = cnt[b];
  v[i] /= (float)(c < 1 ? 1 : c);
}

__global__ void gnorm_apply(float* __restrict__ x, const int* __restrict__ bidx,
                            const float* __restrict__ alpha,
                            const float* __restrict__ gamma,
                            const float* __restrict__ beta,
                            const float* __restrict__ mean,
                            const float* __restrict__ var, long NC, int C)
{
  long i = (long)blockIdx.x * blockDim.x + threadIdx.x;
  if (i >= NC) return;
  int row = (int)(i / C);
  int c = (int)(i - (long)row * C);
  int b = bidx[row];
  float sub = x[i] - alpha[c] * mean[b * C + c];
  x[i] = gamma[c] * sub * rsqrtf(var[b * C + c] + 1e-5f) + beta[c];
}

__global__ void sanitize(float* __restrict__ x, long n)
{
  long i = (long)blockIdx.x * blockDim.x + threadIdx.x;
  if (i >= n) return;
  float v = x[i];
  if (__builtin_isnan(v)) v = 0.f;
  else if (__builtin_isinf(v)) v = (v > 0.f) ? 1e6f : -1e6f;
  v = fminf(fmaxf(v, -1e4f), 1e4f);
  x[i] = v;
}

// h = [x3 (256), center - pos3 (3)], row stride 259
__global__ void build_h4(const float* __restrict__ x3, const float* __restrict__ pos3,
                         const float* __restrict__ minp, float* __restrict__ h, int N)
{
  int i = blockIdx.x * blockDim.x + threadIdx.x;
  if (i >= N * 259) return;
  int row = i / 259, c = i - row * 259;
  float v;
  if (c < 256) {
    v = x3[(size_t)row * 256 + c];
  } else {
    int k = c - 256;
    float p = pos3[row * 3 + k];
    float mp = minp[k];
    const float size3 = 2.0f / 16.0f;
    float center = mp + (floorf((p - mp) / size3) + 0.5f) * size3;
    v = center - p;
  }
  h[i] = v;
}

// ---------------------------------------------------------------------------
// Host side
// ---------------------------------------------------------------------------
extern "C" void kernel_launch(void* const* d_in, const int* in_sizes, int n_in,
                              void* d_out, int out_size, void* d_ws, size_t ws_size,
                              hipStream_t stream)
{
  (void)n_in; (void)out_size; (void)ws_size;

  // ---- input layout (setup_inputs dict order; params in sorted-key order) ----
  const float* feature = (const float*)d_in[0];
  const float* pos0    = (const float*)d_in[1];
  const float* pos1    = (const float*)d_in[2];
  const float* pos2    = (const float*)d_in[3];
  const float* pos3    = (const float*)d_in[4];
  const float* min_pos = (const float*)d_in[5];
  // params leaves start at 6:
  //  ar1..ar5 : base 6,18,30,42,54  [bf1,bf2,bg1,bg2,bh1,bh2,wf1,wf2,wg1,wg2,wh1,wh2]
  //  conv1..3 : base 66,74,82      [b1,b2,bf,bo,w1,w2,wf,wo]
  //  conv4    : base 90            [b1,b2,w1,w2]
  //  lin      : base 94            [b,w]
  //  norm1..3 : base 96,99,102     [alpha,beta,gamma]
  const int* cluster0 = (const int*)d_in[105];
  const int* cluster1 = (const int*)d_in[106];
  const int* cluster2 = (const int*)d_in[107];
  const int* edge1    = (const int*)d_in[108];
  const int* edge2    = (const int*)d_in[109];
  const int* edge3    = (const int*)d_in[110];
  const int* batch1   = (const int*)d_in[111];
  const int* batch2   = (const int*)d_in[112];
  const int* batch3   = (const int*)d_in[113];

  const int N0 = in_sizes[1] / 3;
  const int N1 = in_sizes[2] / 3;
  const int N2 = in_sizes[3] / 3;
  const int N3 = in_sizes[4] / 3;
  const int E1 = in_sizes[108] / 2;
  const int E2 = in_sizes[109] / 2;
  const int E3 = in_sizes[110] / 2;

  auto F = [&](int i) { return (const float*)d_in[i]; };
  auto maxl = [](long a, long b) { return a > b ? a : b; };

  // ---- workspace arena ----
  char* wp = (char*)d_ws;
  auto alloc = [&](long bytes) -> void* {
    void* p = (void*)wp;
    wp += (bytes + 255) & ~255L;
    return p;
  };
  float* x1 = (float*)alloc((long)N1 * 64 * 4);
  float* x2 = (float*)alloc((long)N2 * 128 * 4);
  float* x3 = (float*)alloc((long)N3 * 256 * 4);
  const long SAe = maxl((long)N0 * 32,
                   maxl((long)N1 * 64, maxl((long)N2 * 128, (long)N3 * 259)));
  float* bufA = (float*)alloc(SAe * 4);
  float* bufB = (float*)alloc(SAe * 4);
  float* buf3 = (float*)alloc((long)N0 * 3 * 4);   // rel / dp (N0 is largest N)
  const long SSe = maxl((long)N1 * 32, maxl((long)N2 * 64, (long)N3 * 128));
  float* bufS = (float*)alloc(SSe * 4);
  int*   bufCnt = (int*)alloc((long)N1 * 4);
  float* msum = (float*)alloc(512L * 4);
  float* vsum = (float*)alloc(512L * 4);
  int*   mcnt = (int*)alloc(16L);
  float* h4   = (float*)alloc((long)N3 * 259 * 4);

  // ---- launch helpers ----
  auto gemm = [&](const float* X, const float* W, const float* b, const float* R,
                  float* Y, int N, int K, int M, int act) {
    dim3 g((M + 15) / 16, (N + 15) / 16);
    gemm16<<<g, dim3(32), 0, stream>>>(X, W, b, R, Y, N, K, M, act);
  };
  auto fillF = [&](float* p, float v, long n) {
    fill_f32<<<dim3((unsigned)((n + 255) / 256)), dim3(256), 0, stream>>>(p, v, n);
  };
  auto fillI = [&](int* p, int v, long n) {
    fill_i32<<<dim3((unsigned)((n + 255) / 256)), dim3(256), 0, stream>>>(p, v, n);
  };

  auto pconv = [&](const float* xin, const float* posin, const float* posout,
                   const int* idx, int Nin, int Nout, int cin, int cmid, int cout,
                   int base, float* xout) {
    const float *b1 = F(base + 0), *b2 = F(base + 1), *bf = F(base + 2),
                *bo = F(base + 3), *w1 = F(base + 4), *w2 = F(base + 5),
                *wf = F(base + 6), *wo = F(base + 7);
    build_rel<<<dim3((Nin * 3 + 255) / 256), dim3(256), 0, stream>>>(posin, posout, idx, buf3, Nin);
    gemm(buf3, w1, b1, nullptr, bufA, Nin, 3, cmid, ACT_RELU);
    gemm(bufA, w2, b2, nullptr, bufB, Nin, cmid, cmid, ACT_NONE);   // weight-net
    gemm(xin, wf, bf, nullptr, bufA, Nin, cin, cmid, ACT_NONE);     // features
    fillF(bufS, 0.f, (long)Nout * cmid);
    fillI(bufCnt, 0, (long)Nout);
    const long NC = (long)Nin * cmid;
    scatter_hw<<<dim3((unsigned)((NC + 255) / 256)), dim3(256), 0, stream>>>(
        bufA, bufB, idx, bufS, bufCnt, NC, cmid);
    const long MC = (long)Nout * cmid;
    div_cnt<<<dim3((unsigned)((MC + 255) / 256)), dim3(256), 0, stream>>>(
        bufS, bufCnt, MC, cmid);
    gemm(bufS, wo, bo, nullptr, xout, Nout, cmid, cout, ACT_CELU);
  };

  auto ar = [&](float* x, const float* pos, const int* edge, int E, int N, int d,
                int base, bool endRelu) {
    const float *bf1 = F(base + 0), *bf2 = F(base + 1), *bg1 = F(base + 2),
                *bg2 = F(base + 3), *bh1 = F(base + 4), *bh2 = F(base + 5),
                *wf1 = F(base + 6), *wf2 = F(base + 7), *wg1 = F(base + 8),
                *wg2 = F(base + 9), *wh1 = F(base + 10), *wh2 = F(base + 11);
    gemm(x, wh1, bh1, nullptr, bufA, N, d, d, ACT_RELU);
    gemm(bufA, wh2, bh2, nullptr, buf3, N, d, 3, ACT_NONE);          // dp (N,3)
    fillF(bufB, -3.0e38f, (long)N * d);                              // agg init
    const int* esrc = edge;
    const int* etgt = edge + E;
    const int lds = 16 * (d + 4) * 2 * (int)sizeof(float);
    ar_edge<<<dim3((E + 15) / 16), dim3(2 * d), lds, stream>>>(
        x, pos, buf3, esrc, etgt, E, wf1, bf1, wf2, bf2, bufB, d);
    gemm(bufB, wg1, bg1, nullptr, bufA, N, d, d, ACT_RELU);
    gemm(bufA, wg2, bg2, x, x, N, d, d, endRelu ? ACT_RELU : ACT_NONE);
  };

  auto gnorm = [&](float* x, const int* bidx, int N, int C, int base) {
    const float* alpha = F(base + 0);
    const float* beta  = F(base + 1);
    const float* gamma = F(base + 2);
    fillF(msum, 0.f, 2L * C);
    fillF(vsum, 0.f, 2L * C);
    fillI(mcnt, 0, 2L);
    long nb = ((long)N * C + 255) / 256;
    if (nb > 2048) nb = 2048;
    gnorm_accum<<<dim3((unsigned)nb), dim3(256), 0, stream>>>(
        x, bidx, nullptr, nullptr, msum, mcnt, N, C, 0);
    gnorm_finalize<<<dim3(2), dim3(256), 0, stream>>>(msum, mcnt, C);
    gnorm_accum<<<dim3((unsigned)nb), dim3(256), 0, stream>>>(
        x, bidx, alpha, msum, vsum, mcnt, N, C, 1);
    gnorm_finalize<<<dim3(2), dim3(256), 0, stream>>>(vsum, mcnt, C);
    const long NC = (long)N * C;
    gnorm_apply<<<dim3((unsigned)((NC + 255) / 256)), dim3(256), 0, stream>>>(
        x, bidx, alpha, gamma, beta, msum, vsum, NC, C);
  };

  // ---- forward pass ----
  pconv(feature, pos0, pos1, cluster0, N0, N1, 32, 32, 64, 66, x1);
  ar(x1, pos1, edge1, E1, N1, 64, 6, true);
  gnorm(x1, batch1, N1, 64, 96);

  pconv(x1, pos1, pos2, cluster1, N1, N2, 64, 64, 128, 74, x2);
  ar(x2, pos2, edge2, E2, N2, 128, 18, true);
  gnorm(x2, batch2, N2, 128, 99);

  pconv(x2, pos2, pos3, cluster2, N2, N3, 128, 128, 256, 82, x3);
  ar(x3, pos3, edge3, E3, N3, 256, 30, false);
  ar(x3, pos3, edge3, E3, N3, 256, 42, false);
  ar(x3, pos3, edge3, E3, N3, 256, 54, false);

  sanitize<<<dim3((unsigned)(((long)N3 * 256 + 255) / 256)), dim3(256), 0, stream>>>(
      x3, (long)N3 * 256);
  gnorm(x3, batch3, N3, 256, 102);

  build_h4<<<dim3((N3 * 259 + 255) / 256), dim3(256), 0, stream>>>(x3, pos3, min_pos, h4, N3);
  gemm(h4,   F(92), F(90), nullptr, bufA, N3, 259, 256, ACT_RELU);   // conv4 layer 1
  gemm(bufA, F(93), F(91), nullptr, bufB, N3, 256, 256, ACT_NONE);   // conv4 layer 2
  gemm(bufB, F(95), F(94), nullptr, (float*)d_out, N3, 256, 9, ACT_NONE); // lin
}